// RNNModel_57380763074874
// MI455X (gfx1250) — compile-verified
//
#include <hip/hip_runtime.h>
#include <cstddef>

typedef __attribute__((ext_vector_type(2))) float v2f;
typedef __attribute__((ext_vector_type(8))) float v8f;

#define BSZ 64
#define TSZ 256
#define FSZ 256
#define HSZ 1024
#define OSZ 256

#define NWAVE 8           // K-split factor per block (256 threads)

// Accumulate one 16x16 f32 tile: C += A(16 x [kBeg,kEnd)) * B([kBeg,kEnd) x 16)
// A row-major, stride lda. Bsrc row-major [N_total, K]; B[k][n] = Bsrc[n][k], so
// a B "column" is a contiguous Bsrc row -> float2 loads, no transpose needed.
// Fragment layout per CDNA5 ISA 7.12.2 (wave32):
//   A: lane l holds A[l&15][k + 2*(l>>4) + {0,1}]  (v2f)
//   B: lane l holds B[k + 2*(l>>4) + {0,1}][l&15]  (v2f)
//   C: lane l, vgpr j  <->  (m = j + 8*(l>>4), n = l&15)
__device__ __forceinline__ void wmma_tile_f32(const float* __restrict__ A0, int lda,
                                              const float* __restrict__ B0, int ldb,
                                              int kBeg, int kEnd, int lane, v8f& acc)
{
    const int rc = lane & 15;          // row of A / col of B handled by this lane
    const int kh = (lane >> 4) << 1;   // lanes 16..31 hold K+2, K+3
    const float* ap = A0 + (size_t)rc * lda + kh;
    const float* bp = B0 + (size_t)rc * ldb + kh;
#pragma unroll 4
    for (int k = kBeg; k < kEnd; k += 4) {
        v2f a = *(const v2f*)(ap + k);
        v2f b = *(const v2f*)(bp + k);
        acc = __builtin_amdgcn_wmma_f32_16x16x4_f32(false, a, false, b,
                                                    (short)0, acc, false, false);
    }
}

// One recurrence step, fused input projection:
//   hnext = tanh( x[:,t,:] @ W_ih^T + b_ih + b_hh + hprev @ W_hh^T )
// grid = (4, 64) tiles of 16x16 over [B=64, H=1024]; block = 256 threads (8 waves).
// K (256 + 1024) is split 8 ways across waves; partials reduced through LDS.
__global__ void __launch_bounds__(32 * NWAVE)
rnn_step(const float* __restrict__ x,
         const float* __restrict__ W_ih,
         const float* __restrict__ W_hh,
         const float* __restrict__ b_ih,
         const float* __restrict__ b_hh,
         const float* __restrict__ hprev,
         float* __restrict__ hnext,
         int t, int has_h)
{
    __shared__ float red[NWAVE * 32 * 8];
    const int lane = threadIdx.x & 31;
    const int wave = threadIdx.x >> 5;
    const int tm = blockIdx.x;   // 0..3   (batch tiles)
    const int tn = blockIdx.y;   // 0..63  (hidden tiles)

    v8f acc = {};

    // Phase 1: input projection slice  A = x[tm*16.., t, :] (lda = T*F), B = W_ih
    {
        const float* A0 = x + (size_t)(tm * 16) * (TSZ * FSZ) + (size_t)t * FSZ;
        const float* B0 = W_ih + (size_t)(tn * 16) * FSZ;
        wmma_tile_f32(A0, TSZ * FSZ, B0, FSZ,
                      wave * (FSZ / NWAVE), (wave + 1) * (FSZ / NWAVE), lane, acc);
    }
    // Phase 2: recurrence  A = hprev[tm*16.., :], B = W_hh  (skipped at t==0, h0==0)
    if (has_h) {
        const float* A0 = hprev + (size_t)(tm * 16) * HSZ;
        const float* B0 = W_hh + (size_t)(tn * 16) * HSZ;
        wmma_tile_f32(A0, HSZ, B0, HSZ,
                      wave * (HSZ / NWAVE), (wave + 1) * (HSZ / NWAVE), lane, acc);
    }

    // Cross-wave K reduction through LDS
    {
        const int base = wave * 256 + lane * 8;
#pragma unroll
        for (int j = 0; j < 8; ++j) red[base + j] = acc[j];
    }
    __syncthreads();

    if (wave == 0) {
        const int n  = lane & 15;
        const int mh = (lane >> 4) << 3;     // lanes 16..31 hold rows M=8..15
        const int hh = tn * 16 + n;
        const float bias = b_ih[hh] + b_hh[hh];
#pragma unroll
        for (int j = 0; j < 8; ++j) {
            const int idx = lane * 8 + j;
            float s = 0.f;
#pragma unroll
            for (int w = 0; w < NWAVE; ++w) s += red[w * 256 + idx];
            const int bb = tm * 16 + mh + j;
            hnext[(size_t)bb * HSZ + hh] = tanhf(s + bias);
        }
    }
}

// out = h_last @ W_fc^T + b_fc    ([64,1024] x [1024,256]^T -> [64,256])
__global__ void __launch_bounds__(32 * NWAVE)
fc_head(const float* __restrict__ hlast,
        const float* __restrict__ W_fc,
        const float* __restrict__ b_fc,
        float* __restrict__ out)
{
    __shared__ float red[NWAVE * 32 * 8];
    const int lane = threadIdx.x & 31;
    const int wave = threadIdx.x >> 5;
    const int tm = blockIdx.x;   // 0..3
    const int tn = blockIdx.y;   // 0..15

    v8f acc = {};
    {
        const float* A0 = hlast + (size_t)(tm * 16) * HSZ;
        const float* B0 = W_fc + (size_t)(tn * 16) * HSZ;
        wmma_tile_f32(A0, HSZ, B0, HSZ,
                      wave * (HSZ / NWAVE), (wave + 1) * (HSZ / NWAVE), lane, acc);
    }
    {
        const int base = wave * 256 + lane * 8;
#pragma unroll
        for (int j = 0; j < 8; ++j) red[base + j] = acc[j];
    }
    __syncthreads();

    if (wave == 0) {
        const int n  = lane & 15;
        const int mh = (lane >> 4) << 3;
        const int oo = tn * 16 + n;
        const float bias = b_fc[oo];
#pragma unroll
        for (int j = 0; j < 8; ++j) {
            const int idx = lane * 8 + j;
            float s = 0.f;
#pragma unroll
            for (int w = 0; w < NWAVE; ++w) s += red[w * 256 + idx];
            const int bb = tm * 16 + mh + j;
            out[(size_t)bb * OSZ + oo] = s + bias;
        }
    }
}

extern "C" void kernel_launch(void* const* d_in, const int* in_sizes, int n_in,
                              void* d_out, int out_size, void* d_ws, size_t ws_size,
                              hipStream_t stream)
{
    const float* x    = (const float*)d_in[0];
    const float* W_ih = (const float*)d_in[1];
    const float* W_hh = (const float*)d_in[2];
    const float* b_ih = (const float*)d_in[3];
    const float* b_hh = (const float*)d_in[4];
    const float* W_fc = (const float*)d_in[5];
    const float* b_fc = (const float*)d_in[6];
    float* out = (float*)d_out;

    // Ping-pong hidden-state buffers in workspace (2 * 64*1024 floats = 512 KB)
    float* h0 = (float*)d_ws;
    float* h1 = h0 + BSZ * HSZ;
    float* h[2] = { h0, h1 };

    const dim3 sgrid(BSZ / 16, HSZ / 16);   // (4, 64)
    for (int t = 0; t < TSZ; ++t) {
        const float* hp = h[t & 1];         // ignored at t==0
        float*       hn = h[(t + 1) & 1];
        rnn_step<<<sgrid, 32 * NWAVE, 0, stream>>>(x, W_ih, W_hh, b_ih, b_hh,
                                                   hp, hn, t, (t > 0) ? 1 : 0);
    }
    const float* hlast = h[TSZ & 1];        // TSZ even -> h[0]
    fc_head<<<dim3(BSZ / 16, OSZ / 16), 32 * NWAVE, 0, stream>>>(hlast, W_fc, b_fc, out);
}